// GNNEncoder_81518479278731
// MI455X (gfx1250) — compile-verified
//
#include <hip/hip_runtime.h>
#include <math.h>

#define NN 100000
#define EE 1600000
#define HH 4
#define CC 32
#define DD 128
#define GG 64
#define LL 4
#define INDIM 7
#define KPAD 8
#define NEG_SLOPE 0.2f

typedef __attribute__((ext_vector_type(2))) float v2f;
typedef __attribute__((ext_vector_type(8))) float v8f;

__device__ __forceinline__ float lrelu(float t) {
    return t > 0.f ? t : NEG_SLOPE * t;
}

// ---------------------------------------------------------------------------
// Zero-padding helpers for layer 0 so the GEMM is branch-free.
// xp[N x 8] = x[N x 7] | 0 ; Wp[8 x 128] = W0[7 x 128] | 0
// ---------------------------------------------------------------------------
__global__ void k_pad_x(const float* __restrict__ x, float* __restrict__ xp) {
    int t = blockIdx.x * blockDim.x + threadIdx.x;
    if (t >= NN * KPAD) return;
    const int n = t >> 3, c = t & 7;
    xp[t] = (c < INDIM) ? x[n * INDIM + c] : 0.f;
}

__global__ void k_pad_w(const float* __restrict__ W0, float* __restrict__ Wp) {
    int t = blockIdx.x * blockDim.x + threadIdx.x;
    if (t >= KPAD * DD) return;
    Wp[t] = (t < INDIM * DD) ? W0[t] : 0.f;
}

// ---------------------------------------------------------------------------
// WMMA fp32 GEMM: C[N x 128] = A[N x K] @ B[K x 128], K compile-time mult of 4.
// One wave computes one 16x16 tile with V_WMMA_F32_16X16X4_F32; 8 waves/block
// cover the 8 column tiles of D=128; blockIdx.x indexes 16-row tiles.
// All loads unconditional -> no exec-mask churn, EXEC stays all-ones.
// ---------------------------------------------------------------------------
template <int K>
__global__ __launch_bounds__(256) void k_gemm(const float* __restrict__ A,
                                              const float* __restrict__ B,
                                              float* __restrict__ Cm) {
    const int lane = threadIdx.x & 31;
    const int wave = threadIdx.x >> 5;
    const int row0 = blockIdx.x * 16;
    const int col0 = wave * 16;
    const int m    = lane & 15;  // row within A tile / col within B,C tile
    const int half = lane >> 4;  // K-pair selector per ISA layout

    const float* __restrict__ arow = A + (row0 + m) * K + half * 2;
    const float* __restrict__ bcol = B + col0 + m;

    v8f acc = {};
#pragma unroll 8
    for (int k0 = 0; k0 < K; k0 += 4) {
        const int ka = k0 + half * 2;
        // A tile 16x4: lanes 0-15 hold K=0,1 ; lanes 16-31 hold K=2,3
        const v2f a = *(const v2f*)(arow + k0);  // 8B-aligned global_load_b64
        v2f b;  // B tile 4x16 (row-major K x 128): symmetric striping
        b.x = bcol[ka * DD];
        b.y = bcol[(ka + 1) * DD];
        acc = __builtin_amdgcn_wmma_f32_16x16x4_f32(
            /*neg_a=*/false, a, /*neg_b=*/false, b,
            /*c_mod=*/(short)0, acc, /*reuse_a=*/false, /*reuse_b=*/false);
    }
    // C/D layout: VGPR r -> M=r (lanes 0-15) or M=r+8 (lanes 16-31); N = lane&15
    float* out = Cm + row0 * DD + col0;
#pragma unroll
    for (int r = 0; r < 8; ++r) {
        out[(r + half * 8) * DD + m] = acc[r];
    }
}

// ---------------------------------------------------------------------------
// Per (node, head) attention coefficients: alpha_src/dst = <h[n,h,:], a[h,:]>
// ---------------------------------------------------------------------------
__global__ void k_alpha(const float* __restrict__ hLin,
                        const float* __restrict__ a_src,
                        const float* __restrict__ a_dst,
                        float* __restrict__ aS, float* __restrict__ aD) {
    int t = blockIdx.x * blockDim.x + threadIdx.x;
    if (t >= NN * HH) return;
    const int n = t >> 2, hh = t & 3;
    const float4* hp = (const float4*)(hLin + n * DD + hh * CC);
    const float4* as = (const float4*)(a_src + hh * CC);
    const float4* ad = (const float4*)(a_dst + hh * CC);
    float s = 0.f, d = 0.f;
#pragma unroll
    for (int j = 0; j < CC / 4; ++j) {
        float4 v = hp[j], pa = as[j], pb = ad[j];
        s += v.x * pa.x + v.y * pa.y + v.z * pa.z + v.w * pa.w;
        d += v.x * pb.x + v.y * pb.y + v.z * pb.z + v.w * pb.w;
    }
    aS[t] = s;
    aD[t] = d;
}

// zero aggregation buffer, init segment-max to -inf-ish, zero softmax denom
__global__ void k_init(float* __restrict__ agg, float* __restrict__ mb,
                       float* __restrict__ zb) {
    int t = blockIdx.x * blockDim.x + threadIdx.x;
    if (t < NN * DD) agg[t] = 0.f;
    if (t < NN * HH) { mb[t] = -3.0e38f; zb[t] = 0.f; }
}

// ---------------------------------------------------------------------------
// Edge pass 1: segment max of leaky-relu logits (4 heads per thread)
// ---------------------------------------------------------------------------
__global__ void k_edge_max(const int* __restrict__ ei,
                           const float* __restrict__ aS,
                           const float* __restrict__ aD,
                           float* __restrict__ mb) {
    int e = blockIdx.x * blockDim.x + threadIdx.x;
    if (e >= EE + NN) return;
    int s, d;
    if (e < EE) { s = ei[e]; d = ei[EE + e]; } else { s = d = e - EE; }
    const float4 as = *(const float4*)(aS + s * HH);
    const float4 ad = *(const float4*)(aD + d * HH);
    float* mp = mb + d * HH;
    atomicMax(mp + 0, lrelu(as.x + ad.x));
    atomicMax(mp + 1, lrelu(as.y + ad.y));
    atomicMax(mp + 2, lrelu(as.z + ad.z));
    atomicMax(mp + 3, lrelu(as.w + ad.w));
}

// Edge pass 2: z[dst] += exp(logit - m[dst])
__global__ void k_edge_sum(const int* __restrict__ ei,
                           const float* __restrict__ aS,
                           const float* __restrict__ aD,
                           const float* __restrict__ mb,
                           float* __restrict__ zb) {
    int e = blockIdx.x * blockDim.x + threadIdx.x;
    if (e >= EE + NN) return;
    int s, d;
    if (e < EE) { s = ei[e]; d = ei[EE + e]; } else { s = d = e - EE; }
    const float4 as = *(const float4*)(aS + s * HH);
    const float4 ad = *(const float4*)(aD + d * HH);
    const float4 mv = *(const float4*)(mb + d * HH);
    float* zp = zb + d * HH;
    atomicAdd(zp + 0, expf(lrelu(as.x + ad.x) - mv.x));
    atomicAdd(zp + 1, expf(lrelu(as.y + ad.y) - mv.y));
    atomicAdd(zp + 2, expf(lrelu(as.z + ad.z) - mv.z));
    atomicAdd(zp + 3, expf(lrelu(as.w + ad.w) - mv.w));
}

// Edge pass 3: one thread per (edge, head): agg[dst] += alpha * h_lin[src]
__global__ void k_edge_agg(const int* __restrict__ ei,
                           const float* __restrict__ aS,
                           const float* __restrict__ aD,
                           const float* __restrict__ mb,
                           const float* __restrict__ zb,
                           const float* __restrict__ hLin,
                           float* __restrict__ agg) {
    int t = blockIdx.x * blockDim.x + threadIdx.x;
    if (t >= (EE + NN) * HH) return;
    const int e = t >> 2, hh = t & 3;
    int s, d;
    if (e < EE) { s = ei[e]; d = ei[EE + e]; } else { s = d = e - EE; }
    const float lg = lrelu(aS[s * HH + hh] + aD[d * HH + hh]);
    const float alpha = expf(lg - mb[d * HH + hh]) / (zb[d * HH + hh] + 1e-16f);
    const float4* hp = (const float4*)(hLin + s * DD + hh * CC);
    float* op = agg + d * DD + hh * CC;
#pragma unroll
    for (int j = 0; j < CC / 4; ++j) {
        float4 v = hp[j];
        atomicAdd(op + 4 * j + 0, v.x * alpha);
        atomicAdd(op + 4 * j + 1, v.y * alpha);
        atomicAdd(op + 4 * j + 2, v.z * alpha);
        atomicAdd(op + 4 * j + 3, v.w * alpha);
    }
}

// ---------------------------------------------------------------------------
// Post: one wave32 per node -> bias + LayerNorm (shfl reductions) + ELU + res.
// Safe in-place on `agg` (each wave reads its row fully before writing it).
// ---------------------------------------------------------------------------
__global__ __launch_bounds__(256) void k_post(const float* __restrict__ agg,
                                              const float* __restrict__ bias,
                                              const float* __restrict__ gma,
                                              const float* __restrict__ bta,
                                              const float* __restrict__ resid,
                                              float* __restrict__ outp,
                                              int has_res) {
    const int wid  = (blockIdx.x * blockDim.x + threadIdx.x) >> 5;
    const int lane = threadIdx.x & 31;
    if (wid >= NN) return;
    const float* ag = agg + wid * DD;
    float v[4];
    float sum = 0.f;
#pragma unroll
    for (int j = 0; j < 4; ++j) {
        const int c = lane + 32 * j;
        v[j] = ag[c] + bias[c];
        sum += v[j];
    }
#pragma unroll
    for (int off = 16; off > 0; off >>= 1) sum += __shfl_xor(sum, off, 32);
    const float mu = sum * (1.f / 128.f);
    float var = 0.f;
#pragma unroll
    for (int j = 0; j < 4; ++j) { float dl = v[j] - mu; var += dl * dl; }
#pragma unroll
    for (int off = 16; off > 0; off >>= 1) var += __shfl_xor(var, off, 32);
    const float rs = rsqrtf(var * (1.f / 128.f) + 1e-5f);
#pragma unroll
    for (int j = 0; j < 4; ++j) {
        const int c = lane + 32 * j;
        float y = (v[j] - mu) * rs * gma[c] + bta[c];
        y = y > 0.f ? y : expm1f(y);  // ELU
        outp[wid * DD + c] = (has_res ? resid[wid * DD + c] : 0.f) + y;
    }
}

// ---------------------------------------------------------------------------
// Mean pooling per graph
// ---------------------------------------------------------------------------
__global__ void k_zero_pool(float* __restrict__ gsum, float* __restrict__ gcnt) {
    int t = blockIdx.x * blockDim.x + threadIdx.x;
    if (t < GG * DD) gsum[t] = 0.f;
    if (t < GG) gcnt[t] = 0.f;
}

__global__ void k_pool_sum(const float* __restrict__ h,
                           const int* __restrict__ batch,
                           float* __restrict__ gsum, float* __restrict__ gcnt) {
    int t = blockIdx.x * blockDim.x + threadIdx.x;
    if (t >= NN * DD) return;
    const int n = t >> 7, c = t & 127;
    const int g = batch[n];
    atomicAdd(gsum + g * DD + c, h[t]);
    if (c == 0) atomicAdd(gcnt + g, 1.f);
}

__global__ void k_pool_div(const float* __restrict__ gsum,
                           const float* __restrict__ gcnt,
                           float* __restrict__ outg) {
    int t = blockIdx.x * blockDim.x + threadIdx.x;
    if (t >= GG * DD) return;
    outg[t] = gsum[t] / fmaxf(gcnt[t >> 7], 1.f);
}

// ---------------------------------------------------------------------------
extern "C" void kernel_launch(void* const* d_in, const int* in_sizes, int n_in,
                              void* d_out, int out_size, void* d_ws, size_t ws_size,
                              hipStream_t stream) {
    (void)in_sizes; (void)n_in; (void)out_size; (void)ws_size;

    const float* x     = (const float*)d_in[0];
    const int*   ei    = (const int*)d_in[1];
    const int*   batch = (const int*)d_in[2];
    const float* W0    = (const float*)d_in[3];
    const float* as0   = (const float*)d_in[4];
    const float* ad0   = (const float*)d_in[5];
    const float* b0    = (const float*)d_in[6];
    const float* Wsl   = (const float*)d_in[7];
    const float* asrcs = (const float*)d_in[8];
    const float* adsts = (const float*)d_in[9];
    const float* bsl   = (const float*)d_in[10];
    const float* lng   = (const float*)d_in[11];
    const float* lnb   = (const float*)d_in[12];
    float* out = (float*)d_out;

    float* ws   = (float*)d_ws;
    float* A0v  = ws;           ws += NN * DD;    // ping
    float* A1v  = ws;           ws += NN * DD;    // pong
    float* Lin  = ws;           ws += NN * DD;    // h @ W
    float* xp   = ws;           ws += NN * KPAD;  // padded x (layer 0)
    float* Wp   = ws;           ws += KPAD * DD;  // padded W0 (layer 0)
    float* aS   = ws;           ws += NN * HH;
    float* aD   = ws;           ws += NN * HH;
    float* mb   = ws;           ws += NN * HH;
    float* zb   = ws;           ws += NN * HH;
    float* gsum = ws;           ws += GG * DD;
    float* gcnt = ws;           ws += GG;

    const int nTot = EE + NN;

    // one-time (per call) zero-padding for the layer-0 GEMM
    k_pad_x<<<(NN * KPAD + 255) / 256, 256, 0, stream>>>(x, xp);
    k_pad_w<<<(KPAD * DD + 255) / 256, 256, 0, stream>>>(W0, Wp);

    for (int layer = 0; layer < LL; ++layer) {
        const float* Ain;  float* agg;  const float* resid;
        switch (layer) {
            case 0: Ain = xp;  agg = A0v; resid = nullptr; break;
            case 1: Ain = A0v; agg = A1v; resid = A0v;     break;
            case 2: Ain = A1v; agg = A0v; resid = A1v;     break;
            default:Ain = A0v; agg = out; resid = A0v;     break; // last -> d_out
        }
        const float* asv  = (layer == 0) ? as0 : asrcs + (layer - 1) * HH * CC;
        const float* adv  = (layer == 0) ? ad0 : adsts + (layer - 1) * HH * CC;
        const float* bias = (layer == 0) ? b0  : bsl   + (layer - 1) * DD;

        if (layer == 0) {
            k_gemm<KPAD><<<NN / 16, 256, 0, stream>>>(Ain, Wp, Lin);
        } else {
            k_gemm<DD><<<NN / 16, 256, 0, stream>>>(
                Ain, Wsl + (layer - 1) * DD * DD, Lin);
        }
        k_alpha<<<(NN * HH + 255) / 256, 256, 0, stream>>>(Lin, asv, adv, aS, aD);
        k_init<<<(NN * DD + 255) / 256, 256, 0, stream>>>(agg, mb, zb);
        k_edge_max<<<(nTot + 255) / 256, 256, 0, stream>>>(ei, aS, aD, mb);
        k_edge_sum<<<(nTot + 255) / 256, 256, 0, stream>>>(ei, aS, aD, mb, zb);
        k_edge_agg<<<(nTot * HH + 255) / 256, 256, 0, stream>>>(ei, aS, aD, mb, zb, Lin, agg);
        k_post<<<(NN * 32 + 255) / 256, 256, 0, stream>>>(
            agg, bias, lng + layer * DD, lnb + layer * DD, resid, agg,
            layer > 0 ? 1 : 0);
    }

    k_zero_pool<<<(GG * DD + 255) / 256, 256, 0, stream>>>(gsum, gcnt);
    k_pool_sum<<<(NN * DD + 255) / 256, 256, 0, stream>>>(out, batch, gsum, gcnt);
    k_pool_div<<<(GG * DD + 255) / 256, 256, 0, stream>>>(gsum, gcnt, out + NN * DD);
}